// GIN_Node_44272522887301
// MI455X (gfx1250) — compile-verified
//
#include <hip/hip_runtime.h>

typedef float v2f __attribute__((ext_vector_type(2)));
typedef float v8f __attribute__((ext_vector_type(8)));

#define DCH 64
#define AS  68   // padded LDS A-row stride (floats): conflict-free paired b64 reads
#define WS  80   // W section stride in PAIRS: (80*2)%64==32 -> halves hit disjoint banks

// ---------------------------------------------------------------- utilities
__global__ __launch_bounds__(256) void zero_buf_kernel(float* __restrict__ p, int count) {
  int i = blockIdx.x * blockDim.x + threadIdx.x;
  if (i < count) p[i] = 0.0f;
}

__global__ __launch_bounds__(256) void copy_kernel(const float* __restrict__ s,
                                                   float* __restrict__ d, int count) {
  int i = blockIdx.x * blockDim.x + threadIdx.x;
  if (i < count) d[i] = s[i];
}

// GIN aggregation (eps=0): h[dst] += x[src] per edge. 16 threads/edge, 4 ch each.
__global__ __launch_bounds__(256) void scatter_add_kernel(const float* __restrict__ x,
                                                          const int* __restrict__ srcIdx,
                                                          const int* __restrict__ dstIdx,
                                                          float* __restrict__ h, int e) {
  int t = blockIdx.x * blockDim.x + threadIdx.x;
  int edge = t >> 4;
  if (edge >= e) return;
  int cg = (t & 15) << 2;
  int s = srcIdx[edge];
  int d = dstIdx[edge];
  const float* xs = x + (size_t)s * DCH + cg;
  float* hd = h + (size_t)d * DCH + cg;
  atomicAdd(hd + 0, xs[0]);
  atomicAdd(hd + 1, xs[1]);
  atomicAdd(hd + 2, xs[2]);
  atomicAdd(hd + 3, xs[3]);
}

// ------------------------------------------------- fused GEMM+ReLU+BN-stats
// out = relu( BN_prev(in) @ Wk + bk ), accumulating per-channel sum / sumsq.
// 256 threads = 8 waves; each wave owns a 16-row strip x all 64 output cols
// (4 WMMA column tiles). WMMA: V_WMMA_F32_16X16X4_F32, K-loop of 16 steps.
// W is staged in LDS in fragment-paired layout: section s = k0/2 holds
// {W[2s][n], W[2s+1][n]} contiguously so each B fragment is one ds_load_b64.
__global__ __launch_bounds__(256)
void gemm64_relu_stats(const float* __restrict__ in,    // n x 64
                       const float* __restrict__ Wk,    // 64 x 64 [in][out]
                       const float* __restrict__ bk,    // 64
                       const float* __restrict__ mu,    // 64 (prev sublayer BN)
                       const float* __restrict__ rstd,  // 64
                       const float* __restrict__ gam,   // 64
                       const float* __restrict__ bet,   // 64
                       int applyBN,
                       float* __restrict__ out,         // n x 64
                       float* __restrict__ gsum,        // 64
                       float* __restrict__ gsumsq,      // 64
                       int n) {
  __shared__ __align__(16) float sW2[32 * WS * 2];      // 32 K-sections of 64(+pad) pairs
  __shared__ __align__(16) float sA[8][16 * AS];
  __shared__ float sSum[DCH];
  __shared__ float sSumSq[DCH];

  const int tid  = threadIdx.x;
  const int lane = tid & 31;
  const int wave = tid >> 5;
  const int m     = lane & 15;      // row within A tile / col within B tile
  const int khalf = lane >> 4;      // which K-pair of the 4-wide K block
  const int rowBase = blockIdx.x * 128 + wave * 16;

  if (tid < DCH) { sSum[tid] = 0.0f; sSumSq[tid] = 0.0f; }

  // stage Wk (64x64) into LDS in paired layout
#pragma unroll
  for (int t = 0; t < 4; ++t) {
    int idx = t * 256 + tid;        // float4 index, 1024 total
    int r  = idx >> 4;              // K row 0..63
    int c4 = (idx & 15) << 2;       // out-col group
    float4 w = *(const float4*)(Wk + r * DCH + c4);
    float* dstp = sW2 + ((r >> 1) * WS) * 2 + (r & 1);
    dstp[(c4 + 0) * 2] = w.x;
    dstp[(c4 + 1) * 2] = w.y;
    dstp[(c4 + 2) * 2] = w.z;
    dstp[(c4 + 3) * 2] = w.w;
  }

  // stage this wave's 16x64 A strip, folding in the previous sublayer's BN
#pragma unroll
  for (int t = 0; t < 8; ++t) {
    int idx = t * 32 + lane;        // float4 index within strip (256 total)
    int r  = idx >> 4;
    int c4 = (idx & 15) << 2;
    int row = rowBase + r;
    float4 v = make_float4(0.0f, 0.0f, 0.0f, 0.0f);
    if (row < n) {
      v = *(const float4*)(in + (size_t)row * DCH + c4);
      if (applyBN) {
        v.x = (v.x - mu[c4 + 0]) * rstd[c4 + 0] * gam[c4 + 0] + bet[c4 + 0];
        v.y = (v.y - mu[c4 + 1]) * rstd[c4 + 1] * gam[c4 + 1] + bet[c4 + 1];
        v.z = (v.z - mu[c4 + 2]) * rstd[c4 + 2] * gam[c4 + 2] + bet[c4 + 2];
        v.w = (v.w - mu[c4 + 3]) * rstd[c4 + 3] * gam[c4 + 3] + bet[c4 + 3];
      }
    }
    *(float4*)(&sA[wave][r * AS + c4]) = v;
  }

  __syncthreads();

  v8f acc[4] = {};
  const float* Arow = &sA[wave][m * AS] + khalf * 2;   // + kk*4 per step
  const float* Brow = sW2 + (khalf * WS + m) * 2;      // + kk*2*WS*2 + ct*32 per frag

#pragma unroll
  for (int kk = 0; kk < 16; ++kk) {
    v2f a = *(const v2f*)(Arow + kk * 4);
    const float* bp = Brow + kk * (2 * WS * 2);
#pragma unroll
    for (int ct = 0; ct < 4; ++ct) {
      v2f bfrag = *(const v2f*)(bp + ct * 32);
      acc[ct] = __builtin_amdgcn_wmma_f32_16x16x4_f32(
          false, a, false, bfrag, (short)0, acc[ct], false, false);
    }
  }

  // epilogue: bias + relu + store + per-channel stat partials
  const int rOff = khalf * 8;
#pragma unroll
  for (int ct = 0; ct < 4; ++ct) {
    const int col = ct * 16 + m;
    const float bias = bk[col];
    float s = 0.0f, sq = 0.0f;
#pragma unroll
    for (int i = 0; i < 8; ++i) {
      int row = rowBase + rOff + i;
      float v = fmaxf(acc[ct][i] + bias, 0.0f);
      if (row < n) {
        out[(size_t)row * DCH + col] = v;
        s += v;
        sq += v * v;
      }
    }
    atomicAdd(&sSum[col], s);
    atomicAdd(&sSumSq[col], sq);
  }

  __syncthreads();
  if (tid < DCH) {
    atomicAdd(&gsum[tid], sSum[tid]);
    atomicAdd(&gsumsq[tid], sSumSq[tid]);
  }
}

// compute mu / rstd from accumulated stats; reset accumulators for next use
__global__ __launch_bounds__(64) void bn_finalize_kernel(float* __restrict__ gsum,
                                                         float* __restrict__ gsumsq,
                                                         float* __restrict__ mu,
                                                         float* __restrict__ rstd,
                                                         float invN) {
  int c = threadIdx.x;
  float s = gsum[c];
  float sq = gsumsq[c];
  float mean = s * invN;
  float var = sq * invN - mean * mean;
  mu[c] = mean;
  rstd[c] = rsqrtf(var + 1e-5f);
  gsum[c] = 0.0f;
  gsumsq[c] = 0.0f;
}

// x_out = relu( BN(h) )   (after last sublayer of each GIN layer)
__global__ __launch_bounds__(256) void bn_relu_kernel(const float* __restrict__ h,
                                                      const float* __restrict__ mu,
                                                      const float* __restrict__ rstd,
                                                      const float* __restrict__ gam,
                                                      const float* __restrict__ bet,
                                                      float* __restrict__ xout, int count) {
  int i = blockIdx.x * blockDim.x + threadIdx.x;
  if (i < count) {
    int c = i & (DCH - 1);
    float v = (h[i] - mu[c]) * rstd[c] * gam[c] + bet[c];
    xout[i] = fmaxf(v, 0.0f);
  }
}

// ------------------------------------------------------ classifier (64->16)
__global__ __launch_bounds__(256)
void classifier_gemm(const float* __restrict__ x, const float* __restrict__ cw,
                     const float* __restrict__ cb, float* __restrict__ out, int n) {
  __shared__ __align__(16) float sW2[32 * 16 * 2];   // paired layout, 16 pairs/section
  __shared__ __align__(16) float sA[8][16 * AS];

  const int tid  = threadIdx.x;
  const int lane = tid & 31;
  const int wave = tid >> 5;
  const int m     = lane & 15;
  const int khalf = lane >> 4;
  const int rowBase = blockIdx.x * 128 + wave * 16;

  {
    int r  = tid >> 2;             // K row 0..63
    int c4 = (tid & 3) << 2;       // out-col group
    float4 w = *(const float4*)(cw + r * 16 + c4);
    float* dstp = sW2 + ((r >> 1) * 16) * 2 + (r & 1);
    dstp[(c4 + 0) * 2] = w.x;
    dstp[(c4 + 1) * 2] = w.y;
    dstp[(c4 + 2) * 2] = w.z;
    dstp[(c4 + 3) * 2] = w.w;
  }

#pragma unroll
  for (int t = 0; t < 8; ++t) {
    int idx = t * 32 + lane;
    int r  = idx >> 4;
    int c4 = (idx & 15) << 2;
    int row = rowBase + r;
    float4 v = make_float4(0.0f, 0.0f, 0.0f, 0.0f);
    if (row < n) v = *(const float4*)(x + (size_t)row * DCH + c4);
    *(float4*)(&sA[wave][r * AS + c4]) = v;
  }

  __syncthreads();

  v8f acc = {};
  const float* Arow = &sA[wave][m * AS] + khalf * 2;
  const float* Brow = sW2 + (khalf * 16 + m) * 2;
#pragma unroll
  for (int kk = 0; kk < 16; ++kk) {
    v2f a = *(const v2f*)(Arow + kk * 4);
    v2f bfrag = *(const v2f*)(Brow + kk * 64);
    acc = __builtin_amdgcn_wmma_f32_16x16x4_f32(
        false, a, false, bfrag, (short)0, acc, false, false);
  }

  const float bias = cb[m];
  const int rOff = khalf * 8;
#pragma unroll
  for (int i = 0; i < 8; ++i) {
    int row = rowBase + rOff + i;
    if (row < n) out[(size_t)row * 16 + m] = acc[i] + bias;
  }
}

// ---------------------------------------------------------------- launcher
extern "C" void kernel_launch(void* const* d_in, const int* in_sizes, int n_in,
                              void* d_out, int out_size, void* d_ws, size_t ws_size,
                              hipStream_t stream) {
  (void)n_in; (void)out_size; (void)ws_size;

  const float* x_in   = (const float*)d_in[0];
  const int*   eidx   = (const int*)d_in[1];
  const float* W      = (const float*)d_in[2];
  const float* b      = (const float*)d_in[3];
  const float* gamma  = (const float*)d_in[4];
  const float* beta   = (const float*)d_in[5];
  const float* cls_W  = (const float*)d_in[6];
  const float* cls_b  = (const float*)d_in[7];

  const int N = in_sizes[0] / DCH;                 // 50000
  const int E = in_sizes[1] / 2;                   // 800000
  const int L = in_sizes[2] / (DCH * DCH * DCH);   // 2

  const int* srcIdx = eidx;
  const int* dstIdx = eidx + E;

  const int N64 = N * DCH;
  float* xbuf   = (float*)d_ws;
  float* h0     = xbuf + (size_t)N64;
  float* h1     = h0 + (size_t)N64;
  float* gsum   = h1 + (size_t)N64;
  float* gsumsq = gsum + DCH;
  float* mu     = gsumsq + DCH;
  float* rstd   = mu + DCH;

  const int gemmGrid = (N + 127) / 128;
  const float invN = 1.0f / (float)N;

  // zero the stat accumulators (ws is poisoned by harness)
  zero_buf_kernel<<<1, 256, 0, stream>>>(gsum, 2 * DCH);

  for (int l = 0; l < L; ++l) {
    const float* xcur = (l == 0) ? x_in : xbuf;

    // h0 = x + segment_sum(x[src], dst)
    copy_kernel<<<(N64 + 255) / 256, 256, 0, stream>>>(xcur, h0, N64);
    scatter_add_kernel<<<((size_t)E * 16 + 255) / 256, 256, 0, stream>>>(
        xcur, srcIdx, dstIdx, h0, E);

    float* hin  = h0;
    float* hout = h1;
    for (int k = 0; k < DCH; ++k) {
      const size_t li  = (size_t)l * DCH + k;
      const size_t lip = (size_t)l * DCH + (k > 0 ? k - 1 : 0);
      gemm64_relu_stats<<<gemmGrid, 256, 0, stream>>>(
          hin, W + li * DCH * DCH, b + li * DCH,
          mu, rstd, gamma + lip * DCH, beta + lip * DCH,
          (k > 0) ? 1 : 0,
          hout, gsum, gsumsq, N);
      bn_finalize_kernel<<<1, 64, 0, stream>>>(gsum, gsumsq, mu, rstd, invN);
      float* tmp = hin; hin = hout; hout = tmp;
    }
    // hin now holds h_63; mu/rstd hold its BN stats
    const size_t lilast = (size_t)l * DCH + (DCH - 1);
    bn_relu_kernel<<<(N64 + 255) / 256, 256, 0, stream>>>(
        hin, mu, rstd, gamma + lilast * DCH, beta + lilast * DCH, xbuf, N64);
  }

  classifier_gemm<<<gemmGrid, 256, 0, stream>>>(xbuf, cls_W, cls_b, (float*)d_out, N);
}